// MultiLayerHGNN_65652870087174
// MI455X (gfx1250) — compile-verified
//
#include <hip/hip_runtime.h>

typedef float v2f __attribute__((ext_vector_type(2)));
typedef float v8f __attribute__((ext_vector_type(8)));
typedef int   v4i_gcc __attribute__((vector_size(16)));
typedef __attribute__((address_space(1))) v4i_gcc* as1_v4i;
typedef __attribute__((address_space(3))) v4i_gcc* as3_v4i;

#define HD 128          // hidden dim D
#define LDS_PITCH 132   // padded LDS row pitch (floats): conflict-free strided reads
#define MROWS 64        // output rows per block (4 x 16-row WMMA tiles)

#if defined(__gfx1250__) && __has_builtin(__builtin_amdgcn_global_load_async_to_lds_b128)
#define HGNN_ASYNC_LDS 1
#endif

__device__ __forceinline__ void hgnn_wait_async() {
#ifdef HGNN_ASYNC_LDS
#if __has_builtin(__builtin_amdgcn_s_wait_asynccnt)
  __builtin_amdgcn_s_wait_asynccnt(0);
#else
  asm volatile("s_wait_asynccnt 0" ::: "memory");
#endif
#endif
}

// ---------------------------------------------------------------------------
// GEMM: out[m,:] = ((rsin?rsin[m]:1) * A[m,:]) @ W + bias) * (rsout?rsout[m]:1)
// W (128x128, row-major) staged in LDS via CDNA5 async global->LDS copy
// (ASYNCcnt-tracked, no VGPR round-trip); 64 A-rows staged per block.
// 8 waves x 16 cols; each wave holds 4 accumulators (4 M tiles) and shares one
// B fragment per K-step across 4 V_WMMA_F32_16X16X4_F32.
// Safe in place (out == A): a block reads only its own 64 A-rows into LDS
// (before the barrier) and writes only those rows.
// ---------------------------------------------------------------------------
__global__ __launch_bounds__(256) void hgnn_gemm_kernel(
    const float* __restrict__ A, const float* __restrict__ W,
    const float* __restrict__ bias,
    const float* __restrict__ rsin, const float* __restrict__ rsout,
    float* __restrict__ out, int M)
{
  extern __shared__ float smem[];
  float* Ws = smem;                       // 128 x LDS_PITCH
  float* Xs = smem + HD * LDS_PITCH;      // MROWS x LDS_PITCH

  const int t  = threadIdx.x;
  const int m0 = blockIdx.x * MROWS;

  // ---- Stage W (128x128 = 4096 float4) into padded LDS ----
#pragma unroll
  for (int j = 0; j < 16; ++j) {
    int fidx = t + 256 * j;
    int row  = fidx >> 5;                 // 32 float4 per row
    int c4   = fidx & 31;
    float* lptr       = Ws + row * LDS_PITCH + c4 * 4;
    const float* gptr = W + fidx * 4;
#ifdef HGNN_ASYNC_LDS
    // Dynamic LDS starts at offset 0 of the workgroup allocation; AS(3)
    // addresses are byte offsets into LDS.
    unsigned loff = (unsigned)((const char*)lptr - (const char*)smem);
    __builtin_amdgcn_global_load_async_to_lds_b128(
        (as1_v4i)(unsigned long long)(const void*)gptr,
        (as3_v4i)(unsigned long long)loff, 0, 0);
#else
    *(float4*)lptr = *(const float4*)gptr;
#endif
  }

  // ---- Stage 64 rows of A (2048 float4) with optional per-row input scale ----
#pragma unroll
  for (int j = 0; j < 8; ++j) {
    int fidx = t + 256 * j;
    int row  = fidx >> 5;
    int c4   = fidx & 31;
    int gr   = m0 + row;
    float4 v = make_float4(0.f, 0.f, 0.f, 0.f);
    if (gr < M) {
      v = ((const float4*)(A + (size_t)gr * HD))[c4];
      float sc = rsin ? rsin[gr] : 1.0f;
      v.x *= sc; v.y *= sc; v.z *= sc; v.w *= sc;
    }
    ((float4*)(Xs + row * LDS_PITCH))[c4] = v;
  }

  hgnn_wait_async();
  __syncthreads();

  const int lane    = t & 31;
  const int wave    = t >> 5;
  const int colBase = wave * 16;     // 8 waves cover 128 output columns
  const int half    = lane >> 4;     // 0 or 1
  const int m       = lane & 15;

  // ISA layouts (CDNA5 ISA §7.12.2):
  //  A 16x4 f32:  vgpr0 = K=(0|2), vgpr1 = K=(1|3); lanes 0-15 -> K 0/1, 16-31 -> K 2/3
  //  C/D 16x16:   row = 8*half + vgpr, col = lane&15
  v8f acc[4];
#pragma unroll
  for (int mt = 0; mt < 4; ++mt)
    acc[mt] = (v8f){0.f, 0.f, 0.f, 0.f, 0.f, 0.f, 0.f, 0.f};

#pragma unroll 4
  for (int k = 0; k < HD; k += 4) {
    const int kk = k + half * 2;
    v2f b;
    b[0] = Ws[(kk)     * LDS_PITCH + colBase + m];
    b[1] = Ws[(kk + 1) * LDS_PITCH + colBase + m];
#pragma unroll
    for (int mt = 0; mt < 4; ++mt) {
      const int r = mt * 16 + m;
      v2f a;
      a[0] = Xs[r * LDS_PITCH + kk];
      a[1] = Xs[r * LDS_PITCH + kk + 1];
      acc[mt] = __builtin_amdgcn_wmma_f32_16x16x4_f32(false, a, false, b,
                                                      (short)0, acc[mt],
                                                      false, false);
    }
  }

  const int col    = colBase + m;
  const float bcol = bias[col];
#pragma unroll
  for (int mt = 0; mt < 4; ++mt) {
#pragma unroll
    for (int v = 0; v < 8; ++v) {
      int row = m0 + mt * 16 + half * 8 + v;
      if (row < M) {
        float so = rsout ? rsout[row] : 1.0f;
        out[(size_t)row * HD + col] = (acc[mt][v] + bcol) * so;
      }
    }
  }
}

// ---------------------------------------------------------------------------
// Edge scatter: dst[scatter_idx[e], :] += src[gather_idx[e], :]
// One wave32 per edge, float4 per lane (32*4 = 128 floats).
// ---------------------------------------------------------------------------
__global__ __launch_bounds__(256) void hgnn_scatter_kernel(
    const float* __restrict__ src, const int* __restrict__ gather_idx,
    const int* __restrict__ scatter_idx, float* __restrict__ dst, int nE)
{
  int e    = blockIdx.x * 8 + (threadIdx.x >> 5);
  int lane = threadIdx.x & 31;
  if (e >= nE) return;
  int s = gather_idx[e];
  int d = scatter_idx[e];
  float4 v = ((const float4*)(src + (size_t)s * HD))[lane];
  float* drow = dst + (size_t)d * HD + lane * 4;
  atomicAdd(drow + 0, v.x);
  atomicAdd(drow + 1, v.y);
  atomicAdd(drow + 2, v.z);
  atomicAdd(drow + 3, v.w);
}

// Degree counts (float accumulators so we can reuse buffers as reciprocals).
__global__ __launch_bounds__(256) void hgnn_count_kernel(
    const int* __restrict__ node_idx, const int* __restrict__ he_idx,
    float* __restrict__ node_cnt, float* __restrict__ he_cnt, int nE)
{
  int e = blockIdx.x * 256 + threadIdx.x;
  if (e >= nE) return;
  atomicAdd(&node_cnt[node_idx[e]], 1.0f);
  atomicAdd(&he_cnt[he_idx[e]], 1.0f);
}

// node_rden = 1/max(cnt,1)   (scatter_mean clamp);  he_rden = 1/(cnt + 1e-8)
__global__ __launch_bounds__(256) void hgnn_recip_kernel(
    float* __restrict__ node_rden, int n, float* __restrict__ he_rden, int h)
{
  int i = blockIdx.x * 256 + threadIdx.x;
  if (i < n) node_rden[i] = 1.0f / fmaxf(node_rden[i], 1.0f);
  if (i < h) he_rden[i]   = 1.0f / (he_rden[i] + 1e-8f);
}

__device__ __forceinline__ float wave_sum32(float v) {
#pragma unroll
  for (int off = 16; off > 0; off >>= 1) v += __shfl_xor(v, off, 32);
  return v;
}

// ---------------------------------------------------------------------------
// Fused epilogue per node row (one wave32 per row, float4 per lane):
//   v = accum*rden + xt;  v = LN(v; g1,b1);  v = leaky_relu(v,0.2);
//   v = LN(v; g2,b2);  x_io = v (+ x_io if residual)
// ---------------------------------------------------------------------------
__global__ __launch_bounds__(256) void hgnn_fuse_kernel(
    const float* __restrict__ accum, const float* __restrict__ rden,
    const float* __restrict__ xt,
    const float* __restrict__ g1, const float* __restrict__ b1,
    const float* __restrict__ g2, const float* __restrict__ b2,
    float* __restrict__ x_io, int nRows, int residual)
{
  int row  = blockIdx.x * 8 + (threadIdx.x >> 5);
  int lane = threadIdx.x & 31;
  if (row >= nRows) return;

  const float rd = rden[row];
  float4 a  = ((const float4*)(accum + (size_t)row * HD))[lane];
  float4 tv = ((const float4*)(xt    + (size_t)row * HD))[lane];
  float vx = a.x * rd + tv.x;
  float vy = a.y * rd + tv.y;
  float vz = a.z * rd + tv.z;
  float vw = a.w * rd + tv.w;

  // LayerNorm 1
  float mu = wave_sum32(vx + vy + vz + vw) * (1.0f / HD);
  float dx = vx - mu, dy = vy - mu, dz = vz - mu, dw = vw - mu;
  float var = wave_sum32(dx * dx + dy * dy + dz * dz + dw * dw) * (1.0f / HD);
  float rs = rsqrtf(var + 1e-5f);
  float4 gg = ((const float4*)g1)[lane];
  float4 bb = ((const float4*)b1)[lane];
  vx = dx * rs * gg.x + bb.x;
  vy = dy * rs * gg.y + bb.y;
  vz = dz * rs * gg.z + bb.z;
  vw = dw * rs * gg.w + bb.w;

  // LeakyReLU(0.2)
  vx = vx >= 0.f ? vx : 0.2f * vx;
  vy = vy >= 0.f ? vy : 0.2f * vy;
  vz = vz >= 0.f ? vz : 0.2f * vz;
  vw = vw >= 0.f ? vw : 0.2f * vw;

  // LayerNorm 2
  mu = wave_sum32(vx + vy + vz + vw) * (1.0f / HD);
  dx = vx - mu; dy = vy - mu; dz = vz - mu; dw = vw - mu;
  var = wave_sum32(dx * dx + dy * dy + dz * dz + dw * dw) * (1.0f / HD);
  rs = rsqrtf(var + 1e-5f);
  gg = ((const float4*)g2)[lane];
  bb = ((const float4*)b2)[lane];
  vx = dx * rs * gg.x + bb.x;
  vy = dy * rs * gg.y + bb.y;
  vz = dz * rs * gg.z + bb.z;
  vw = dw * rs * gg.w + bb.w;

  if (residual) {
    float4 xp = ((const float4*)(x_io + (size_t)row * HD))[lane];
    vx += xp.x; vy += xp.y; vz += xp.z; vw += xp.w;
  }
  ((float4*)(x_io + (size_t)row * HD))[lane] = make_float4(vx, vy, vz, vw);
}

extern "C" void kernel_launch(void* const* d_in, const int* in_sizes, int n_in,
                              void* d_out, int out_size, void* d_ws, size_t ws_size,
                              hipStream_t stream)
{
  const float* x    = (const float*)d_in[0];
  const int*   hei  = (const int*)d_in[1];   // (2, E): [node_idx | he_idx]
  const float* hw   = (const float*)d_in[2]; // (H,)
  const float* Wn   = (const float*)d_in[3]; // (L, D, D)
  const float* bn   = (const float*)d_in[4]; // (L, D)
  const float* We   = (const float*)d_in[5];
  const float* be   = (const float*)d_in[6];
  const float* lig  = (const float*)d_in[7];
  const float* lib  = (const float*)d_in[8];
  const float* logq = (const float*)d_in[9];
  const float* lob  = (const float*)d_in[10];

  const int N = in_sizes[0] / HD;
  const int E = in_sizes[1] / 2;
  const int H = in_sizes[2];
  const int L = in_sizes[4] / HD;

  const int* node_idx = hei;
  const int* he_idx   = hei + E;

  // Workspace layout (floats): xt | node_accum | he_buf | node_rden | he_rden
  float* xt         = (float*)d_ws;
  float* node_accum = xt + (size_t)N * HD;
  float* he_buf     = node_accum + (size_t)N * HD;
  float* node_rden  = he_buf + (size_t)H * HD;
  float* he_rden    = node_rden + N;

  float* xout = (float*)d_out;

  const size_t shmem = (size_t)(HD + MROWS) * LDS_PITCH * sizeof(float); // ~99 KB

  // Degree counts + reciprocals (invariant across layers).
  (void)hipMemsetAsync(node_rden, 0, (size_t)(N + H) * sizeof(float), stream);
  hgnn_count_kernel<<<(E + 255) / 256, 256, 0, stream>>>(node_idx, he_idx,
                                                         node_rden, he_rden, E);
  int mx = N > H ? N : H;
  hgnn_recip_kernel<<<(mx + 255) / 256, 256, 0, stream>>>(node_rden, N,
                                                          he_rden, H);

  for (int i = 0; i < L; ++i) {
    const float* xin = (i == 0) ? x : (const float*)xout;

    // xt = xin @ Wn[i] + bn[i]
    hgnn_gemm_kernel<<<(N + MROWS - 1) / MROWS, 256, shmem, stream>>>(
        xin, Wn + (size_t)i * HD * HD, bn + (size_t)i * HD,
        nullptr, nullptr, xt, N);

    // nodes -> hyperedges (sum); mean folded into next GEMM's rsin
    (void)hipMemsetAsync(he_buf, 0, (size_t)H * HD * sizeof(float), stream);
    hgnn_scatter_kernel<<<(E + 7) / 8, 256, 0, stream>>>(xt, node_idx, he_idx,
                                                         he_buf, E);

    // he = ((he/denom) @ We[i] + be[i]) * hyperedge_weight   (in place)
    hgnn_gemm_kernel<<<(H + MROWS - 1) / MROWS, 256, shmem, stream>>>(
        he_buf, We + (size_t)i * HD * HD, be + (size_t)i * HD,
        he_rden, hw, he_buf, H);

    // hyperedges -> nodes (sum); mean folded into fuse via node_rden
    (void)hipMemsetAsync(node_accum, 0, (size_t)N * HD * sizeof(float), stream);
    hgnn_scatter_kernel<<<(E + 7) / 8, 256, 0, stream>>>(he_buf, he_idx,
                                                         node_idx, node_accum, E);

    // LN -> LeakyReLU -> LN (+ residual for i>0), written into x buffer
    hgnn_fuse_kernel<<<(N + 7) / 8, 256, 0, stream>>>(
        node_accum, node_rden, xt,
        lig + (size_t)i * HD, lib + (size_t)i * HD,
        logq + (size_t)i * HD, lob + (size_t)i * HD,
        xout, N, i > 0);
  }
}